// EulerWithEchoModel_75892072121092
// MI455X (gfx1250) — compile-verified
//
#include <hip/hip_runtime.h>
#include <math.h>

// ---------------- problem constants (from reference) ----------------
#define BB 16
#define SS 2048
#define VV 4096
#define DD 32           // D
#define HH 4            // heads
#define DH 8            // D/H
#define KK 64           // 2*D, GEMM K
#define LUT_N 4096
#define TWO_PI_F 6.2831853071795864769f
#define INV_TWO_PI_F 0.15915494309189533577f
#define PHI_F 1.6180339887498948482f

typedef float v2f __attribute__((ext_vector_type(2)));
typedef float v8f __attribute__((ext_vector_type(8)));

// =====================================================================
// Kernel 1: sequential Euler/echo scan. 16 blocks (one per batch),
// 32 threads (one wave, one lane per dim d). Produces combs[S][B][64].
// =====================================================================
__global__ __launch_bounds__(32)
void euler_echo_scan(const int* __restrict__ ids,
                     const float* __restrict__ emb,       // [V, 64]
                     const float* __restrict__ W_gate,    // [64, 4]
                     const float* __restrict__ b_gate,    // [4]
                     const float* __restrict__ decay_logit,// [4]
                     const float* __restrict__ echo_scale, // [1]
                     float* __restrict__ combs)           // [S,B,64]
{
    __shared__ __align__(16) float2 lut[LUT_N];   // (sin, cos) pairs, 32KB
    __shared__ int sids[SS];                      // 8KB

    const int b = blockIdx.x;       // batch
    const int d = threadIdx.x;      // dim 0..31

    // Build LUT (sin, cos) with linear-interp lookup semantics of reference
    for (int i = d; i < LUT_N; i += 32) {
        float ang = (float)i * (TWO_PI_F / (float)LUT_N);
        float sn, cs;
        sincosf(ang, &sn, &cs);
        lut[i] = make_float2(sn, cs);
    }
    // Stage this batch's token ids
    for (int i = d; i < SS; i += 32) sids[i] = ids[b * SS + i];
    __syncthreads();

    // Per-lane constants
    const float4 wg_lo = *(const float4*)&W_gate[d * 4];         // W_gate[d][0..3]
    const float4 wg_hi = *(const float4*)&W_gate[(DD + d) * 4];  // W_gate[32+d][0..3]
    const int   hd  = d >> 3;                                    // head of this dim
    const float bg  = b_gate[hd];
    const float dlg = decay_logit[hd];
    const float dec = 1.0f / (1.0f + __expf(-dlg));              // sigmoid
    const float es  = echo_scale[0];

    float h_r = 0.f, h_i = 0.f, m_r = 0.f, m_i = 0.f;

    // Depth-4 software pipeline for the embedding gathers
    const int PF = 4;
    float wq[4], bq[4];
#pragma unroll
    for (int k = 0; k < PF; ++k) {
        int id = sids[k];
        wq[k] = emb[id * KK + d];
        bq[k] = emb[id * KK + DD + d];
    }

    for (int s0 = 0; s0 < SS; s0 += PF) {
#pragma unroll
        for (int u = 0; u < PF; ++u) {
            const int s = s0 + u;
            const float w_t = wq[u];
            const float b_t = bq[u];
            const int sp = s + PF;
            if (sp < SS) {                     // prefetch 4 steps ahead
                int idn = sids[sp];
                wq[u] = emb[idn * KK + d];
                bq[u] = emb[idn * KK + DD + d];
            }

            // theta = h/(1+|w|) + b_t + s*PHI
            const float rw    = 1.0f / (1.0f + fabsf(w_t));
            const float phase = b_t + (float)s * PHI_F;
            const float th_r  = fmaf(h_r, rw, phase);
            const float th_i  = fmaf(h_i, rw, phase);

            // LUT sin/cos with linear interpolation
            float u_r = th_r * INV_TWO_PI_F; u_r -= floorf(u_r);
            float u_i = th_i * INV_TWO_PI_F; u_i -= floorf(u_i);
            float pos_r = u_r * (float)LUT_N;
            float pos_i = u_i * (float)LUT_N;
            float i0fr = floorf(pos_r), i0fi = floorf(pos_i);
            float fr = pos_r - i0fr,    fi = pos_i - i0fi;
            int i0r = ((int)i0fr) & (LUT_N - 1);
            int i0i = ((int)i0fi) & (LUT_N - 1);
            int i1r = (i0r + 1) & (LUT_N - 1);
            int i1i = (i0i + 1) & (LUT_N - 1);
            float2 e0r = lut[i0r], e1r = lut[i1r];
            float2 e0i = lut[i0i], e1i = lut[i1i];
            float s_r = fmaf(fr, e1r.x - e0r.x, e0r.x);
            float c_r = fmaf(fr, e1r.y - e0r.y, e0r.y);
            float s_i = fmaf(fi, e1i.x - e0i.x, e0i.x);
            float c_i = fmaf(fi, e1i.y - e0i.y, e0i.y);

            // complex rotation
            const float nh_r = c_r * c_i - s_r * s_i;
            const float nh_i = c_r * s_i + s_r * c_i;
            h_r = nh_r; h_i = nh_i;

            // gate = sigmoid(concat(h_r,h_i) @ W_gate + b_gate)  (wave reduce)
            float p0 = fmaf(h_r, wg_lo.x, h_i * wg_hi.x);
            float p1 = fmaf(h_r, wg_lo.y, h_i * wg_hi.y);
            float p2 = fmaf(h_r, wg_lo.z, h_i * wg_hi.z);
            float p3 = fmaf(h_r, wg_lo.w, h_i * wg_hi.w);
#pragma unroll
            for (int off = 16; off >= 1; off >>= 1) {
                p0 += __shfl_xor(p0, off, 32);
                p1 += __shfl_xor(p1, off, 32);
                p2 += __shfl_xor(p2, off, 32);
                p3 += __shfl_xor(p3, off, 32);
            }
            float psel = (hd == 0) ? p0 : (hd == 1) ? p1 : (hd == 2) ? p2 : p3;
            const float g = 1.0f / (1.0f + __expf(-(psel + bg)));

            // echo memory + output
            m_r = fmaf(dec, m_r, g * h_r);
            m_i = fmaf(dec, m_i, g * h_i);
            const float out_r = fmaf(es, m_r, h_r);
            const float out_i = fmaf(es, m_i, h_i);

            float* crow = combs + ((size_t)s * BB + b) * KK;
            crow[d]      = out_r;
            crow[DD + d] = out_i;
        }
    }
}

// =====================================================================
// Kernel 2: logits = combs[M=S*B, 64] x W_out^T[64, V] + b_out,
// written as out[b, s, v].  WMMA f32 16x16x4, K chained 16x.
// Block: 256 thr = 8 waves. Block owns 128 vocab cols (LDS-staged),
// wave w owns s = blockIdx.y*8 + w  (its 16 M-rows are batches 0..15).
// =====================================================================
#define NW 8            // waves per block
#define NFRAG 8         // 16-col N fragments per wave  -> 128 cols
#define LDS_KSTRIDE 68  // padded row stride (floats) -> conflict-free b64 reads

__global__ __launch_bounds__(256)
void proj_wmma(const float* __restrict__ combs,   // [S,B,64]
               const float* __restrict__ W_out,   // [V,64]
               const float* __restrict__ b_out,   // [V]
               float* __restrict__ out)           // [B,S,V]
{
    __shared__ __align__(16) float ldsB[128 * LDS_KSTRIDE];  // ~34KB

    const int tid  = threadIdx.x;
    const int wave = tid >> 5;
    const int lane = tid & 31;
    const int l16  = lane & 15;
    const int hi   = lane >> 4;          // 0: lanes 0-15, 1: lanes 16-31

    const int v0b = blockIdx.x * 128;    // vocab slab base
    const int s   = blockIdx.y * NW + wave;

    // ---- stage W_out[v0b : v0b+128, 0:64] into padded LDS (coalesced) ----
    {
        const float4* src = (const float4*)(W_out + (size_t)v0b * KK);
#pragma unroll
        for (int i = 0; i < 8; ++i) {
            int idx4 = tid + i * 256;            // float4 index, 0..2047
            int vloc = idx4 >> 4;                // /16 float4 per row
            int k4   = (idx4 & 15) << 2;         // float offset in row
            *(float4*)&ldsB[vloc * LDS_KSTRIDE + k4] = src[idx4];
        }
    }
    __syncthreads();

    // ---- accumulators: 8 fragments of 16x16 ----
    v8f acc[NFRAG];
#pragma unroll
    for (int j = 0; j < NFRAG; ++j) acc[j] = (v8f){0.f,0.f,0.f,0.f,0.f,0.f,0.f,0.f};

    // A-fragment source: rows m = s*16 + (lane&15), koff = 2*(lane>>4)
    const float* arow = combs + ((size_t)s * BB + l16) * KK + (hi << 1);

#pragma unroll
    for (int ki = 0; ki < 16; ++ki) {
        const int k0 = ki * 4;
        const v2f a = *(const v2f*)(arow + k0);          // A[l16][k0 + 2*hi .. +1]
#pragma unroll
        for (int j = 0; j < NFRAG; ++j) {
            const int vloc = j * 16 + l16;
            const v2f bf = *(const v2f*)&ldsB[vloc * LDS_KSTRIDE + k0 + (hi << 1)];
            acc[j] = __builtin_amdgcn_wmma_f32_16x16x4_f32(
                         false, a, false, bf, (short)0, acc[j], false, false);
        }
    }

    // ---- epilogue: + b_out, store to out[b, s, v] (streaming) ----
    // Preload the 8 per-fragment biases (one clause of b32 loads).
    float biasv[NFRAG];
#pragma unroll
    for (int j = 0; j < NFRAG; ++j) biasv[j] = b_out[v0b + j * 16 + l16];

    // e (output row / batch) OUTER, j (column fragment) INNER:
    // the 8 consecutive stores per row cover v0b..v0b+127 contiguously,
    // so the nontemporal write stream forms full cache lines.
    const int brow0 = hi << 3;                           // 0 or 8
    float* obase = out + (size_t)s * VV + v0b + l16;
#pragma unroll
    for (int e = 0; e < 8; ++e) {
        float* prow = obase + (size_t)(brow0 + e) * ((size_t)SS * VV);
#pragma unroll
        for (int j = 0; j < NFRAG; ++j) {
            float val = acc[j][e] + biasv[j];
            __builtin_nontemporal_store(val, prow + j * 16);
        }
    }
}

// =====================================================================
extern "C" void kernel_launch(void* const* d_in, const int* in_sizes, int n_in,
                              void* d_out, int out_size, void* d_ws, size_t ws_size,
                              hipStream_t stream) {
    const int*   input_ids   = (const int*)  d_in[0];
    const float* emb_table   = (const float*)d_in[1];
    const float* W_out       = (const float*)d_in[2];
    const float* b_out       = (const float*)d_in[3];
    const float* W_gate      = (const float*)d_in[4];
    const float* b_gate      = (const float*)d_in[5];
    const float* decay_logit = (const float*)d_in[6];
    const float* echo_scale  = (const float*)d_in[7];
    float* out = (float*)d_out;

    float* combs = (float*)d_ws;   // needs S*B*64*4 = 8 MB of scratch

    euler_echo_scan<<<dim3(BB), dim3(32), 0, stream>>>(
        input_ids, emb_table, W_gate, b_gate, decay_logit, echo_scale, combs);

    proj_wmma<<<dim3(VV / 128, SS / NW), dim3(256), 0, stream>>>(
        combs, W_out, b_out, out);
}